// Single_Scale_RPN_Outputs_65481071395167
// MI455X (gfx1250) — compile-verified
//
#include <hip/hip_runtime.h>
#include <hip/hip_bf16.h>
#include <math.h>

typedef __attribute__((ext_vector_type(2))) float v2f;
typedef __attribute__((ext_vector_type(8))) float v8f;

#define Hh 50
#define Ww 84
#define CIN 512
#define COUT 512
#define NPIX 4200            // 50*84
#define KTOT 4608            // 512*9
#define NANCH 15
#define PRE_N 6000
#define POST_N 1000
#define SORT_N 65536
#define CH 4096              // bitonic local chunk
#define BBOX_CLIP 4.135166556742356f   // log(1000/16)

// output layout (floats)
#define OUT_CLS   0
#define OUT_BBOX  126000     // 2*15*4200
#define OUT_ROIS  630000     // + 2*60*4200
#define OUT_PROBS 640000     // + 2*1000*5

// ---------------- Conv 3x3 (implicit GEMM, f32 WMMA) ----------------
// Block tile 128x128x32, 8 waves, each wave owns a 32(M)x64(N) strip:
// 2 M-frags x 4 N-frags -> per 8 WMMAs only 2 A + 4 B b64 LDS frag loads.
#define BM 128
#define BN 128
#define BK 32
#define LDP 34               // padded LDS row (dwords): even -> 8B aligned b64 frag loads

__global__ __launch_bounds__(256) void conv3x3_wmma(
    const float* __restrict__ x, const float* __restrict__ wgt,
    const float* __restrict__ bias, float* __restrict__ hout)
{
  __shared__ float As[BM][LDP];   // [m][k]   17408 B
  __shared__ float Bs[BN][LDP];   // [n][k]   17408 B (k-contiguous per column)

  const int tid = threadIdx.x;
  const int b   = blockIdx.z;
  const int m0  = blockIdx.y * BM;
  const int n0  = blockIdx.x * BN;
  const float* xb = x + (size_t)b * CIN * NPIX;

  // A (weights) loader mapping: 2 threads per row, 16 floats each
  const int arow = tid >> 1;
  const int acol = (tid & 1) * 16;

  // B (im2col) loader mapping: 128 columns, 2 threads per column (16 k each)
  const int bn  = tid & 127;
  const int bk0 = (tid >> 7) * 16;
  const int ng  = n0 + bn;
  const int hh  = ng / Ww;
  const int ww  = ng - hh * Ww;
  const bool ncol_ok = (ng < NPIX);

  // wave -> C strip mapping: 4 waves in M (32 rows each), 2 waves in N (64 each)
  const int lane   = tid & 31;
  const int wv     = tid >> 5;
  const int mstrip = (wv >> 1) * 32;   // 0,32,64,96
  const int nstrip = (wv & 1) * 64;    // 0 / 64
  const int lane15 = lane & 15;
  const int kb2    = (lane >> 4) << 1; // 0 or 2
  const int ml0    = mstrip + lane15;
  const int ml1    = mstrip + 16 + lane15;

  v8f acc[2][4];
  const v8f vz = {0.f,0.f,0.f,0.f,0.f,0.f,0.f,0.f};
  #pragma unroll
  for (int mf = 0; mf < 2; mf++)
    #pragma unroll
    for (int nf = 0; nf < 4; nf++)
      acc[mf][nf] = vz;

  for (int kt = 0; kt < KTOT; kt += BK) {
    // load A tile (row-major weights, coalesced float4 quads)
    {
      const float* ag = wgt + (size_t)(m0 + arow) * KTOT + kt + acol;
      float4 a0 = *reinterpret_cast<const float4*>(ag);
      float4 a1 = *reinterpret_cast<const float4*>(ag + 4);
      float4 a2 = *reinterpret_cast<const float4*>(ag + 8);
      float4 a3 = *reinterpret_cast<const float4*>(ag + 12);
      As[arow][acol +  0] = a0.x; As[arow][acol +  1] = a0.y;
      As[arow][acol +  2] = a0.z; As[arow][acol +  3] = a0.w;
      As[arow][acol +  4] = a1.x; As[arow][acol +  5] = a1.y;
      As[arow][acol +  6] = a1.z; As[arow][acol +  7] = a1.w;
      As[arow][acol +  8] = a2.x; As[arow][acol +  9] = a2.y;
      As[arow][acol + 10] = a2.z; As[arow][acol + 11] = a2.w;
      As[arow][acol + 12] = a3.x; As[arow][acol + 13] = a3.y;
      As[arow][acol + 14] = a3.z; As[arow][acol + 15] = a3.w;
    }
    // load B tile (im2col gather with zero padding)
    {
      int k  = kt + bk0;
      int ci = k / 9;
      int r  = k - ci * 9;
      #pragma unroll
      for (int kk = 0; kk < 16; kk++) {
        int kh = r / 3;
        int kw = r - kh * 3;
        int yy = hh + kh - 1;
        int xx = ww + kw - 1;
        float v = 0.f;
        if (ncol_ok && (unsigned)yy < (unsigned)Hh && (unsigned)xx < (unsigned)Ww)
          v = xb[(size_t)ci * NPIX + yy * Ww + xx];
        Bs[bn][bk0 + kk] = v;
        r++; if (r == 9) { r = 0; ci++; }
      }
    }
    __syncthreads();

    #pragma unroll
    for (int kk = 0; kk < BK; kk += 4) {
      v2f av0 = *(const v2f*)(&As[ml0][kk + kb2]);
      v2f av1 = *(const v2f*)(&As[ml1][kk + kb2]);
      #pragma unroll
      for (int nf = 0; nf < 4; nf++) {
        v2f bv = *(const v2f*)(&Bs[nstrip + nf * 16 + lane15][kk + kb2]);
        acc[0][nf] = __builtin_amdgcn_wmma_f32_16x16x4_f32(
            false, av0, false, bv, (short)0, acc[0][nf], false, false);
        acc[1][nf] = __builtin_amdgcn_wmma_f32_16x16x4_f32(
            false, av1, false, bv, (short)0, acc[1][nf], false, false);
      }
    }
    __syncthreads();
  }

  // epilogue: bias + ReLU, store h
  float bco[2][8];
  #pragma unroll
  for (int mf = 0; mf < 2; mf++)
    #pragma unroll
    for (int vg = 0; vg < 8; vg++)
      bco[mf][vg] = bias[m0 + mstrip + mf * 16 + vg + 8 * (lane >> 4)];

  #pragma unroll
  for (int mf = 0; mf < 2; mf++) {
    #pragma unroll
    for (int nf = 0; nf < 4; nf++) {
      int ngl = n0 + nstrip + nf * 16 + lane15;
      if (ngl < NPIX) {
        #pragma unroll
        for (int vg = 0; vg < 8; vg++) {
          int co = m0 + mstrip + mf * 16 + vg + 8 * (lane >> 4);
          float v = acc[mf][nf][vg] + bco[mf][vg];
          hout[((size_t)b * COUT + co) * NPIX + ngl] = v > 0.f ? v : 0.f;
        }
      }
    }
  }
}

// ---------------- 1x1 heads + sigmoid + sort-key pack ----------------
__global__ __launch_bounds__(256) void heads_1x1(
    const float* __restrict__ hbuf,
    const float* __restrict__ cw, const float* __restrict__ cb,
    const float* __restrict__ bw, const float* __restrict__ bb,
    float* __restrict__ out, unsigned long long* __restrict__ keys)
{
  const int n = blockIdx.x * 256 + threadIdx.x;
  const int b = blockIdx.y;
  if (n >= NPIX) return;
  const float* hb = hbuf + (size_t)b * COUT * NPIX + n;

  float acc[75];
  #pragma unroll
  for (int m = 0; m < 75; m++) acc[m] = 0.f;

  for (int k = 0; k < CIN; k++) {
    float xv = hb[(size_t)k * NPIX];
    #pragma unroll
    for (int m = 0; m < 15; m++) acc[m] += cw[m * CIN + k] * xv;
    #pragma unroll
    for (int m = 0; m < 60; m++) acc[15 + m] += bw[m * CIN + k] * xv;
  }

  float* cls_out  = out + OUT_CLS  + (size_t)b * 15 * NPIX;
  float* bbox_out = out + OUT_BBOX + (size_t)b * 60 * NPIX;
  unsigned long long* kseg = keys + (size_t)b * SORT_N;

  #pragma unroll
  for (int a = 0; a < 15; a++) {
    float lg = acc[a] + cb[a];
    cls_out[a * NPIX + n] = lg;
    float s = 1.f / (1.f + expf(-lg));         // sigmoid > 0
    unsigned u = __float_as_uint(s);
    unsigned idx = (unsigned)(n * 15 + a);
    kseg[idx] = ((unsigned long long)(~u) << 32) | idx;  // asc sort => desc score, low-idx ties
  }
  #pragma unroll
  for (int m = 0; m < 60; m++)
    bbox_out[m * NPIX + n] = acc[15 + m] + bb[m];
}

// ---------------- bitonic sort (u64 keys, 2 segments of 65536) -------
__global__ void key_init(unsigned long long* keys)
{
  int i = blockIdx.x * 256 + threadIdx.x;
  if (i < 2 * SORT_N) keys[i] = 0xFFFFFFFFFFFFFFFFull;
}

__global__ __launch_bounds__(256) void bitonic_local_sort(unsigned long long* keys)
{
  __shared__ unsigned long long sd[CH];
  const int base = blockIdx.x * CH;
  for (int t = threadIdx.x; t < CH; t += 256) sd[t] = keys[base + t];
  __syncthreads();
  for (int k = 2; k <= CH; k <<= 1) {
    for (int j = k >> 1; j > 0; j >>= 1) {
      for (int idx = threadIdx.x; idx < CH; idx += 256) {
        int l = idx ^ j;
        if (l > idx) {
          int gi = (base + idx) & (SORT_N - 1);
          bool up = (gi & k) == 0;
          unsigned long long a = sd[idx], c = sd[l];
          if ((a > c) == up) { sd[idx] = c; sd[l] = a; }
        }
      }
      __syncthreads();
    }
  }
  for (int t = threadIdx.x; t < CH; t += 256) keys[base + t] = sd[t];
}

__global__ void bitonic_global_stage(unsigned long long* keys, int j, int k)
{
  int gid = blockIdx.x * 256 + threadIdx.x;     // 131072 threads
  int seg = gid >> 16;
  int i   = gid & (SORT_N - 1);
  int l   = i ^ j;
  if (l > i) {
    unsigned long long* base = keys + (size_t)seg * SORT_N;
    unsigned long long a = base[i], c = base[l];
    bool up = (i & k) == 0;
    if ((a > c) == up) { base[i] = c; base[l] = a; }
  }
}

__global__ __launch_bounds__(256) void bitonic_local_cascade(unsigned long long* keys, int k)
{
  __shared__ unsigned long long sd[CH];
  const int base = blockIdx.x * CH;
  for (int t = threadIdx.x; t < CH; t += 256) sd[t] = keys[base + t];
  __syncthreads();
  for (int j = CH >> 1; j > 0; j >>= 1) {
    for (int idx = threadIdx.x; idx < CH; idx += 256) {
      int l = idx ^ j;
      if (l > idx) {
        int gi = (base + idx) & (SORT_N - 1);
        bool up = (gi & k) == 0;
        unsigned long long a = sd[idx], c = sd[l];
        if ((a > c) == up) { sd[idx] = c; sd[l] = a; }
      }
    }
    __syncthreads();
  }
  for (int t = threadIdx.x; t < CH; t += 256) keys[base + t] = sd[t];
}

// ---------------- box decode for top 6000 ----------------
__global__ void decode6k(const unsigned long long* __restrict__ keys,
                         const float* __restrict__ bbox_pred,
                         const float* __restrict__ anchors,
                         const float* __restrict__ iminfo,
                         float4* __restrict__ boxes, float* __restrict__ scs,
                         unsigned* __restrict__ valid)
{
  int i = blockIdx.x * 256 + threadIdx.x;
  int b = blockIdx.y;
  if (i >= PRE_N) return;
  unsigned long long key = keys[(size_t)b * SORT_N + i];
  unsigned idx = (unsigned)key;
  float score = __uint_as_float(~(unsigned)(key >> 32));

  int a = (int)(idx % 15);
  int n = (int)(idx / 15);
  int hh = n / Ww, ww = n - hh * Ww;
  float sx = ww * 16.0f, sy = hh * 16.0f;
  float ax1 = anchors[a * 4 + 0] + sx, ay1 = anchors[a * 4 + 1] + sy;
  float ax2 = anchors[a * 4 + 2] + sx, ay2 = anchors[a * 4 + 3] + sy;

  const float* dp = bbox_pred + (size_t)b * 60 * NPIX;
  float dx = dp[(4 * a + 0) * NPIX + n];
  float dy = dp[(4 * a + 1) * NPIX + n];
  float dw = fminf(dp[(4 * a + 2) * NPIX + n], BBOX_CLIP);
  float dh = fminf(dp[(4 * a + 3) * NPIX + n], BBOX_CLIP);

  float wdt = ax2 - ax1 + 1.f, hgt = ay2 - ay1 + 1.f;
  float cx = ax1 + 0.5f * wdt, cy = ay1 + 0.5f * hgt;
  float pcx = dx * wdt + cx, pcy = dy * hgt + cy;
  float pw = expf(dw) * wdt, ph = expf(dh) * hgt;

  float imh = iminfo[b * 3 + 0], imw = iminfo[b * 3 + 1], iscale = iminfo[b * 3 + 2];
  float x1 = fminf(fmaxf(pcx - 0.5f * pw, 0.f), imw - 1.f);
  float y1 = fminf(fmaxf(pcy - 0.5f * ph, 0.f), imh - 1.f);
  float x2 = fminf(fmaxf(pcx + 0.5f * pw - 1.f, 0.f), imw - 1.f);
  float y2 = fminf(fmaxf(pcy + 0.5f * ph - 1.f, 0.f), imh - 1.f);

  float wS = x2 - x1 + 1.f, hS = y2 - y1 + 1.f;
  float xc = x1 + 0.5f * wS, yc = y1 + 0.5f * hS;
  float ms = 0.0f * iscale;   // MIN_SIZE == 0
  bool v = (wS >= ms) && (hS >= ms) && (xc < imw) && (yc < imh);

  boxes[(size_t)b * PRE_N + i] = make_float4(x1, y1, x2, y2);
  scs[(size_t)b * PRE_N + i]   = score;
  valid[(size_t)b * PRE_N + i] = v ? 1u : 0u;
}

// ---------------- greedy NMS + stable compaction ----------------
__global__ __launch_bounds__(256) void nms_compact(
    const float4* __restrict__ boxes, const float* __restrict__ scs,
    const unsigned* __restrict__ valid,
    float* __restrict__ rois_out, float* __restrict__ probs_out)
{
  const int b = blockIdx.x;
  const int tid = threadIdx.x;
  __shared__ float4 sbox[PRE_N];          // 96 KB (LDS: 320 KB/WGP)
  __shared__ unsigned char keepF[PRE_N];
  __shared__ int s_cnt;

  const float4* bbg = boxes + (size_t)b * PRE_N;
  for (int j = tid; j < PRE_N; j += 256) {
    sbox[j] = bbg[j];
    keepF[j] = (unsigned char)valid[(size_t)b * PRE_N + j];
  }
  if (tid == 0) s_cnt = 0;
  __syncthreads();

  // cache this thread's 24 boxes in registers (constant indices only)
  float rx1[24], ry1[24], rx2[24], ry2[24];
  unsigned kb = 0u;
  #pragma unroll
  for (int kq = 0; kq < 24; kq++) {
    int j = tid + kq * 256;
    if (j < PRE_N) {
      float4 B4 = sbox[j];
      rx1[kq] = B4.x; ry1[kq] = B4.y; rx2[kq] = B4.z; ry2[kq] = B4.w;
      if (keepF[j]) kb |= (1u << kq);
    } else {
      rx1[kq] = 0.f; ry1[kq] = 0.f; rx2[kq] = 0.f; ry2[kq] = 0.f;
    }
  }
  __syncthreads();

  int lastI = PRE_N - 1;
  for (int i = 0; i < PRE_N; i++) {
    bool kept = (keepF[i] != 0);
    if (kept) {
      float4 bi = sbox[i];
      float areai = (bi.z - bi.x + 1.f) * (bi.w - bi.y + 1.f);
      #pragma unroll
      for (int kq = 0; kq < 24; kq++) {
        if (kb & (1u << kq)) {
          int j = tid + kq * 256;
          if (j > i) {
            float xx1 = fmaxf(bi.x, rx1[kq]);
            float yy1 = fmaxf(bi.y, ry1[kq]);
            float xx2 = fminf(bi.z, rx2[kq]);
            float yy2 = fminf(bi.w, ry2[kq]);
            float iw = fmaxf(xx2 - xx1 + 1.f, 0.f);
            float ih = fmaxf(yy2 - yy1 + 1.f, 0.f);
            float inter = iw * ih;
            float areaj = (rx2[kq] - rx1[kq] + 1.f) * (ry2[kq] - ry1[kq] + 1.f);
            if (inter > 0.7f * (areai + areaj - inter)) {
              kb &= ~(1u << kq);
              keepF[j] = 0;
            }
          }
        }
      }
      if (tid == 0) s_cnt++;
    }
    __syncthreads();
    if (s_cnt >= POST_N) { lastI = i; break; }   // first 1000 kept are final
  }
  __syncthreads();

  // stable compaction (wave 0, wave32): kept in order, then suppressed in order
  if (tid < 32) {
    const int lane = tid;
    float* rois = rois_out + (size_t)b * POST_N * 5;
    float* pr   = probs_out + (size_t)b * POST_N;

    int base = 0;
    for (int c = 0; c < PRE_N && base < POST_N; c += 32) {
      int j = c + lane;
      bool f = (j < PRE_N) && (j <= lastI) && keepF[j];
      unsigned long long m = __ballot(f);
      int pos = base + (int)__popcll(m & ((1ull << lane) - 1ull));
      if (f && pos < POST_N) {
        float4 B4 = sbox[j];
        rois[pos * 5 + 0] = (float)b;
        rois[pos * 5 + 1] = B4.x; rois[pos * 5 + 2] = B4.y;
        rois[pos * 5 + 3] = B4.z; rois[pos * 5 + 4] = B4.w;
        pr[pos] = scs[(size_t)b * PRE_N + j];
      }
      base += (int)__popcll(m);
    }
    int keptTot = base < POST_N ? base : POST_N;

    int base2 = keptTot;
    for (int c = 0; c < PRE_N && base2 < POST_N; c += 32) {
      int j = c + lane;
      bool f = (j < PRE_N) && !((j <= lastI) && keepF[j]);
      unsigned long long m = __ballot(f);
      int pos = base2 + (int)__popcll(m & ((1ull << lane) - 1ull));
      if (f && pos < POST_N) {
        float4 B4 = sbox[j];
        rois[pos * 5 + 0] = (float)b;
        rois[pos * 5 + 1] = B4.x; rois[pos * 5 + 2] = B4.y;
        rois[pos * 5 + 3] = B4.z; rois[pos * 5 + 4] = B4.w;
        pr[pos] = -1.0f;
      }
      base2 += (int)__popcll(m);
    }
  }
}

// ---------------- host-side launch ----------------
extern "C" void kernel_launch(void* const* d_in, const int* in_sizes, int n_in,
                              void* d_out, int out_size, void* d_ws, size_t ws_size,
                              hipStream_t stream) {
  const float* x       = (const float*)d_in[0];
  const float* im_info = (const float*)d_in[1];
  const float* conv_w  = (const float*)d_in[2];
  const float* conv_b  = (const float*)d_in[3];
  const float* cls_w   = (const float*)d_in[4];
  const float* cls_b   = (const float*)d_in[5];
  const float* bbox_w  = (const float*)d_in[6];
  const float* bbox_b  = (const float*)d_in[7];
  const float* anchors = (const float*)d_in[8];
  float* out = (float*)d_out;

  // workspace layout (bytes)
  char* ws = (char*)d_ws;
  float*              h_buf  = (float*)(ws + 0);                        // 2*512*4200*4 = 17,203,200
  unsigned long long* keys   = (unsigned long long*)(ws + 17203200);    // 2*65536*8 = 1,048,576
  float4*             boxes  = (float4*)(ws + 18251776);                // 2*6000*16 = 192,000
  float*              sc6k   = (float*)(ws + 18443776);                 // 2*6000*4  = 48,000
  unsigned*           valid  = (unsigned*)(ws + 18491776);              // 2*6000*4  = 48,000

  // 1) 3x3 conv + ReLU (f32 WMMA implicit GEMM), 128x128x32 tiles
  conv3x3_wmma<<<dim3(33, 4, 2), dim3(256), 0, stream>>>(x, conv_w, conv_b, h_buf);

  // 2) sort-key padding init, then 1x1 heads (logits/bbox to d_out, keys to ws)
  key_init<<<512, 256, 0, stream>>>(keys);
  heads_1x1<<<dim3(17, 2), dim3(256), 0, stream>>>(h_buf, cls_w, cls_b, bbox_w, bbox_b, out, keys);

  // 3) bitonic sort, ascending u64 (= descending score, stable tie-break)
  bitonic_local_sort<<<32, 256, 0, stream>>>(keys);
  for (int k = 2 * CH; k <= SORT_N; k <<= 1) {
    for (int j = k >> 1; j >= CH; j >>= 1)
      bitonic_global_stage<<<512, 256, 0, stream>>>(keys, j, k);
    bitonic_local_cascade<<<32, 256, 0, stream>>>(keys, k);
  }

  // 4) decode top 6000 proposals per image
  decode6k<<<dim3(24, 2), dim3(256), 0, stream>>>(keys, out + OUT_BBOX, anchors, im_info,
                                                  boxes, sc6k, valid);

  // 5) greedy NMS + compaction into rois/probs
  nms_compact<<<dim3(2), dim3(256), 0, stream>>>(boxes, sc6k, valid,
                                                 out + OUT_ROIS, out + OUT_PROBS);
}